// private_encoder_26843545600016
// MI455X (gfx1250) — compile-verified
//
#include <hip/hip_runtime.h>

typedef __attribute__((ext_vector_type(2))) float v2f;
typedef __attribute__((ext_vector_type(8))) float v8f;

#define N_NODES  100000
#define N_EDGES  1600000
#define IN_FEATS 256
#define OUT_DIM  128

// ---------------------------------------------------------------------------
// deg[n] = 1.0 (self loop)
__global__ void k_deg_init(float* __restrict__ deg) {
    int i = blockIdx.x * blockDim.x + threadIdx.x;
    if (i < N_NODES) deg[i] = 1.0f;
}

// deg[col[e]] += 1
__global__ void k_deg_count(const long long* __restrict__ ei, float* __restrict__ deg) {
    int e = blockIdx.x * blockDim.x + threadIdx.x;
    if (e < N_EDGES) {
        int c = (int)ei[(size_t)N_EDGES + e];
        unsafeAtomicAdd(&deg[c], 1.0f);   // GLOBAL_ATOMIC_ADD_F32, no return
    }
}

// deg -> deg^{-1/2}   (deg >= 1 always, reference guards collapse)
__global__ void k_deg_rsqrt(float* __restrict__ deg) {
    int i = blockIdx.x * blockDim.x + threadIdx.x;
    if (i < N_NODES) deg[i] = __frsqrt_rn(deg[i]);
}

// ---------------------------------------------------------------------------
// h = x @ W + b using V_WMMA_F32_16X16X4_F32.
// One wave computes one 16(M)x16(N) f32 tile; 8 waves/block cover N=128.
// Grid = N_NODES/16 = 6250 blocks.
__global__ __launch_bounds__(256) void k_linear_wmma(const float* __restrict__ x,
                                                     const float* __restrict__ W,
                                                     const float* __restrict__ b,
                                                     float* __restrict__ h) {
    const int lane  = threadIdx.x & 31;
    const int wave  = threadIdx.x >> 5;      // 0..7 -> N tile
    const int mBase = blockIdx.x * 16;       // node rows (exact multiple)
    const int nBase = wave * 16;
    const int half  = lane >> 4;             // 0: lanes 0-15 (K0,K1), 1: lanes 16-31 (K2,K3)
    const int sub   = lane & 15;

    // A: row M=sub of x, K pair selected by half
    const float* aRow = x + (size_t)(mBase + sub) * IN_FEATS + half * 2;
    // B: column N=nBase+sub of W (row-major [256,128]), K pair selected by half
    const float* bCol = W + (size_t)nBase + sub;

    v8f c = {};
    #pragma unroll 8
    for (int k = 0; k < IN_FEATS; k += 4) {
        v2f a, bb;
        a.x  = aRow[k + 0];
        a.y  = aRow[k + 1];
        bb.x = bCol[(size_t)(k + half * 2 + 0) * OUT_DIM];
        bb.y = bCol[(size_t)(k + half * 2 + 1) * OUT_DIM];
        // D = A*B + C  (fp32, matches reference precision)
        c = __builtin_amdgcn_wmma_f32_16x16x4_f32(false, a, false, bb,
                                                  (short)0, c, false, false);
    }

    const int   n    = nBase + sub;
    const float bias = b[n];
    float* outCol = h + (size_t)mBase * OUT_DIM + n;
    #pragma unroll
    for (int i = 0; i < 8; ++i) {
        int m = i + half * 8;                 // C/D layout: vgpr i -> row i (+8 for hi half)
        outCol[(size_t)m * OUT_DIM] = c[i] + bias;
    }
}

// ---------------------------------------------------------------------------
// out[n] = h[n] * dinv[n]^2   (self-loop term; also initializes poisoned d_out)
__global__ void k_out_self(const float* __restrict__ h, const float* __restrict__ dinv,
                           float* __restrict__ out) {
    int idx = blockIdx.x * blockDim.x + threadIdx.x;
    if (idx < N_NODES * OUT_DIM) {
        int n = idx >> 7;                     // /128
        float d = dinv[n];
        out[idx] = h[idx] * d * d;
    }
}

// One wave per edge: out[col] += h[row] * dinv[row]*dinv[col]
// Each lane moves a float4 (128 floats / 32 lanes). h & out live in L2 (51 MB each).
__global__ __launch_bounds__(256) void k_edge_scatter(const long long* __restrict__ ei,
                                                      const float* __restrict__ h,
                                                      const float* __restrict__ dinv,
                                                      float* __restrict__ out) {
    int tid  = blockIdx.x * blockDim.x + threadIdx.x;
    int e    = tid >> 5;
    int lane = tid & 31;
    if (e >= N_EDGES) return;
    int r = (int)ei[e];
    int c = (int)ei[(size_t)N_EDGES + e];
    float norm = dinv[r] * dinv[c];

    const float4 hv = *(const float4*)(h + (size_t)r * OUT_DIM + lane * 4);
    float* o = out + (size_t)c * OUT_DIM + lane * 4;
    unsafeAtomicAdd(o + 0, hv.x * norm);
    unsafeAtomicAdd(o + 1, hv.y * norm);
    unsafeAtomicAdd(o + 2, hv.z * norm);
    unsafeAtomicAdd(o + 3, hv.w * norm);
}

// ReLU in place (must follow all scatter atomics)
__global__ void k_relu(float* __restrict__ out) {
    int i = blockIdx.x * blockDim.x + threadIdx.x;
    if (i < N_NODES * OUT_DIM) out[i] = fmaxf(out[i], 0.0f);
}

// ---------------------------------------------------------------------------
extern "C" void kernel_launch(void* const* d_in, const int* in_sizes, int n_in,
                              void* d_out, int out_size, void* d_ws, size_t ws_size,
                              hipStream_t stream) {
    const float*     x  = (const float*)d_in[0];
    const long long* ei = (const long long*)d_in[1];   // int64 [2, E]
    const float*     W  = (const float*)d_in[2];
    const float*     b  = (const float*)d_in[3];
    float* out = (float*)d_out;

    // workspace: h [N, 128] then dinv [N]
    float* h    = (float*)d_ws;
    float* dinv = h + (size_t)N_NODES * OUT_DIM;

    const int T = 256;
    k_deg_init   <<<(N_NODES + T - 1) / T, T, 0, stream>>>(dinv);
    k_deg_count  <<<(N_EDGES + T - 1) / T, T, 0, stream>>>(ei, dinv);
    k_deg_rsqrt  <<<(N_NODES + T - 1) / T, T, 0, stream>>>(dinv);

    k_linear_wmma<<<N_NODES / 16, T, 0, stream>>>(x, W, b, h);

    k_out_self   <<<(N_NODES * OUT_DIM + T - 1) / T, T, 0, stream>>>(h, dinv, out);
    k_edge_scatter<<<((size_t)N_EDGES * 32 + T - 1) / T, T, 0, stream>>>(ei, h, dinv, out);
    k_relu       <<<(N_NODES * OUT_DIM + T - 1) / T, T, 0, stream>>>(out);
}